// Weak_loss_33724083208818
// MI455X (gfx1250) — compile-verified
//
#include <hip/hip_runtime.h>
#include <stdint.h>

#define N_ROWS 4096
#define N_COLS 32000
#define N_VEC  (N_COLS / 4)   // 8000 float4 per row
#define BLOCK  1024           // 32 waves (wave32)
#define EPSF   1e-10f

typedef __attribute__((ext_vector_type(2))) float v2f;
typedef __attribute__((ext_vector_type(8))) float v8f;

__device__ __forceinline__ float wave_sum(float v) {
#pragma unroll
  for (int o = 16; o > 0; o >>= 1) v += __shfl_xor(v, o, 32);
  return v;
}

// Block-wide sum, result broadcast to all threads. blockDim.x must be 1024.
__device__ __forceinline__ float block_sum(float v, float* sred) {
  v = wave_sum(v);
  const int lane = threadIdx.x & 31;
  const int wid  = threadIdx.x >> 5;
  if (lane == 0) sred[wid] = v;
  __syncthreads();
  if (wid == 0) {
    float r = sred[lane];          // exactly 32 waves
    r = wave_sum(r);
    if (lane == 0) sred[32] = r;
  }
  __syncthreads();
  float out = sred[32];
  __syncthreads();                 // allow sred reuse
  return out;
}

// One block per row. Stage the 128KB row into LDS with gfx1250 async
// global->LDS copies (single HBM pass), then two LDS passes:
//  pass 1: denom = sum(exp(x)) + eps
//  pass 2: acc  += log1p(eps - p) for p = exp(x)/denom strictly below p_label
__global__ __launch_bounds__(BLOCK) void row_loss_kernel(
    const float* __restrict__ scores, const int* __restrict__ labels,
    float* __restrict__ rowloss) {
  __shared__ float4 srow4[N_VEC];   // 128000 B
  __shared__ float  sred[33];

  const int row = blockIdx.x;
  const int tid = threadIdx.x;
  const float* gsrc = scores + (size_t)row * N_COLS;

  // Async copy: each lane issues B128 global->LDS transfers (ASYNCcnt).
  for (int k = tid; k < N_VEC; k += BLOCK) {
    unsigned lds_off = (unsigned)(uintptr_t)(const void*)(&srow4[k]);
    const float* g = gsrc + 4 * k;
    asm volatile("global_load_async_to_lds_b128 %0, %1, off"
                 :: "v"(lds_off), "v"(g)
                 : "memory");
  }
  asm volatile("s_wait_asynccnt 0x0" ::: "memory");
  __syncthreads();  // each wave waited for its own copies; barrier makes all visible

  // Pass 1: row sum of exp
  float s = 0.0f;
  for (int k = tid; k < N_VEC; k += BLOCK) {
    float4 x = srow4[k];
    s += __expf(x.x) + __expf(x.y) + __expf(x.z) + __expf(x.w);
  }
  const float denom = block_sum(s, sred) + EPSF;
  const float rinv  = 1.0f / denom;
  const float plab  = __expf(((const float*)srow4)[labels[row]]) * rinv;

  // Pass 2: masked log1p accumulation (p_label <= ~2e-3 for N(0,1) scores,
  // so 1+(eps-p) keeps full precision in __logf)
  float acc = 0.0f;
  for (int k = tid; k < N_VEC; k += BLOCK) {
    float4 x = srow4[k];
    float p;
    p = __expf(x.x) * rinv; if (p < plab) acc += __logf(1.0f + (EPSF - p));
    p = __expf(x.y) * rinv; if (p < plab) acc += __logf(1.0f + (EPSF - p));
    p = __expf(x.z) * rinv; if (p < plab) acc += __logf(1.0f + (EPSF - p));
    p = __expf(x.w) * rinv; if (p < plab) acc += __logf(1.0f + (EPSF - p));
  }
  const float tot = block_sum(acc, sred);
  if (tid == 0) rowloss[row] = tot;
}

// Single-wave final reduction of the 4096 row losses via WMMA:
// D = A(16x4) * ones(4x16) + C accumulates 64 values per instruction along K.
// A[m] = {v[2m], v[2m+1], v[32+2m], v[32+2m+1]} per the 32-bit A layout
// (lanes 0-15 hold K=0,1; lanes 16-31 hold K=2,3).
__global__ void final_reduce_kernel(const float* __restrict__ rowloss,
                                    float* __restrict__ out) {
  const int lane = threadIdx.x;  // exactly 32 threads, EXEC all ones
  v8f c = {0.f, 0.f, 0.f, 0.f, 0.f, 0.f, 0.f, 0.f};
  v2f b; b.x = 1.0f; b.y = 1.0f;  // B = ones
  const int base_lane = ((lane >> 4) << 5) | ((lane & 15) << 1);
#pragma unroll 4
  for (int it = 0; it < N_ROWS / 64; ++it) {
    const float* p = rowloss + it * 64 + base_lane;
    v2f a; a.x = p[0]; a.y = p[1];
    c = __builtin_amdgcn_wmma_f32_16x16x4_f32(
        /*neg_a=*/false, a, /*neg_b=*/false, b,
        /*c_mod=*/(short)0, c, /*reuse_a=*/false, /*reuse_b=*/false);
  }
  // D[m][n] identical over n; lanes 0..15 hold M=0..7 in c[0..7],
  // lanes 16..31 hold M=8..15. Sum VGPRs, fold lane16 into lane0.
  float t = c[0] + c[1] + c[2] + c[3] + c[4] + c[5] + c[6] + c[7];
  t += __shfl_down(t, 16, 32);
  if (lane == 0) out[0] = -t * (1.0f / (float)N_ROWS);
}

extern "C" void kernel_launch(void* const* d_in, const int* in_sizes, int n_in,
                              void* d_out, int out_size, void* d_ws, size_t ws_size,
                              hipStream_t stream) {
  const float* scores = (const float*)d_in[0];
  const int*   labels = (const int*)d_in[1];
  float* rowloss = (float*)d_ws;  // needs N_ROWS * 4 = 16 KB of scratch

  row_loss_kernel<<<N_ROWS, BLOCK, 0, stream>>>(scores, labels, rowloss);
  final_reduce_kernel<<<1, 32, 0, stream>>>(rowloss, (float*)d_out);
}